// TransducerLoss_429496729689
// MI455X (gfx1250) — compile-verified
//
#include <hip/hip_runtime.h>
#include <hip/hip_bf16.h>
#include <math.h>

// RNN-T loss: B=32, T=256, U=64 (U1=65 joint rows), V=512, blank=0, NEG=-1e5.
#define Bsz 32
#define Tsz 256
#define Usz 64
#define U1  65
#define Vsz 512
#define NEGV (-100000.0f)

typedef __attribute__((ext_vector_type(2))) float v2f;
typedef __attribute__((ext_vector_type(8))) float v8f;

__device__ __forceinline__ float lae(float a, float b) {
    // logaddexp, overflow-safe (exp argument always <= 0)
    float m = fmaxf(a, b);
    float d = fminf(a, b) - m;
    return m + log1pf(__expf(d));
}

// ---------------------------------------------------------------------------
// Kernel 1: gather blank/label log-probs and build the exclusive prefix array
//   c[b,t,u] (u=0..64) via WMMA f32 16x16x4 triangular matmuls.
// One wave handles 16 (b,t) rows. rowgroups = B*T/16 = 512; 4 waves/block.
// ---------------------------------------------------------------------------
__global__ void __launch_bounds__(128) rnnt_prep(
        const float* __restrict__ lp, const int* __restrict__ targets,
        const int* __restrict__ tlen,
        float* __restrict__ blank, float* __restrict__ cpre) {
    __shared__ float slab[4][16][64];          // [wave][row][u] masked label lp
    const int lane = threadIdx.x & 31;
    const int wid  = threadIdx.x >> 5;
    const int hi   = lane >> 4;                // 0: lanes 0-15, 1: lanes 16-31
    const int lo   = lane & 15;
    const int rg   = blockIdx.x * 4 + wid;     // rowgroup 0..511
    const int r0   = rg * 16;                  // first flat (b,t) row
    const int b    = r0 / Tsz;                 // 16 rows share one b (256%16==0)
    const int t0   = r0 % Tsz;
    const int tl   = tlen[b];

    // Gather masked label log-probs (2 of 512 vocab entries per cell touched).
    for (int e = lane; e < 16 * 64; e += 32) {
        int n = e >> 6, u = e & 63;
        float v = NEGV;
        if (u < tl) {
            int tg = targets[b * Usz + u];
            size_t idx = ((size_t)((b * Tsz + t0 + n) * U1 + u)) * Vsz + tg;
            v = lp[idx];
        }
        slab[wid][n][u] = v;
    }
    // Gather blank log-probs straight to workspace (v = 0).
    for (int e = lane; e < 16 * 65; e += 32) {
        int n = e / 65, u = e % 65;
        size_t row = (size_t)((b * Tsz + t0 + n) * U1 + u);
        blank[row] = lp[row * (size_t)Vsz];
    }
    __syncthreads();

    // c[row][0] = 0 (exclusive prefix start)
    if (hi == 0) cpre[(size_t)(r0 + lo) * U1] = 0.0f;

    // Inclusive prefix of 64 labels per row as 4 chunks of T16 x V16 matmuls.
    // A (16x4 slice of lower-tri): lane -> (M=lo, K=2*hi+{0,1}) per ISA 7.12.2.
    // B (4x16): lane -> (K=2*hi+{0,1}, N=lo). D vgpr r -> (M=r+8*hi, N=lo).
    float offset = 0.0f;
    for (int q = 0; q < 4; ++q) {
        v8f acc = {};
        #pragma unroll
        for (int s = 0; s < 4; ++s) {
            const int j0 = 4 * s + 2 * hi;     // global K column of this slice
            v2f A, Bm;
            A.x = (j0     <= lo) ? 1.0f : 0.0f;   // T16[m][j] = (j <= m)
            A.y = (j0 + 1 <= lo) ? 1.0f : 0.0f;
            Bm.x = slab[wid][lo][16 * q + j0];
            Bm.y = slab[wid][lo][16 * q + j0 + 1];
            acc = __builtin_amdgcn_wmma_f32_16x16x4_f32(
                      false, A, false, Bm, (short)0, acc, false, false);
        }
        // Chunk total = element (M=15, N=lo) = vgpr7 of lane 16+lo.
        float tot = __shfl(acc[7], 16 + lo, 32);
        #pragma unroll
        for (int r = 0; r < 8; ++r) {
            int m = r + 8 * hi;                // chunk-local position
            int p = 16 * q + m;                // inclusive position 0..63
            cpre[(size_t)(r0 + lo) * U1 + 1 + p] = acc[r] + offset;
        }
        offset += tot;
    }
}

// ---------------------------------------------------------------------------
// Kernel 2: the sequential t-recurrence. One block (64 thr = 2 wave32) per b.
//   alpha_t = c_t + cumlogsumexp(alpha_{t-1} + blank_{t-1} - c_t)
// u=tid handles 0..63; u=64 is a scalar tail on thread 63.
// ---------------------------------------------------------------------------
__global__ void __launch_bounds__(64) rnnt_alpha(
        const float* __restrict__ blank, const float* __restrict__ cpre,
        const int* __restrict__ ilen, const int* __restrict__ tlen,
        float* __restrict__ ll) {
    __shared__ float sh[64];
    __shared__ float sh_a64;
    const int b = blockIdx.x;
    const int u = threadIdx.x;
    const int il = ilen[b], tl = tlen[b];
    const int t_idx = il - 1;
    const size_t base = (size_t)b * Tsz * U1;

    float alpha = cpre[base + u];              // alpha0 = c[:,0,:]
    if (u == 63) sh_a64 = cpre[base + 64];
    __syncthreads();
    float a64 = sh_a64;

    if (t_idx == 0) {                          // not hit for this data, but exact
        if (tl < 64 && u == tl) ll[b] = alpha + blank[base + tl];
        if (tl == 64 && u == 63) ll[b] = a64 + blank[base + 64];
    }

    for (int t = 1; t < Tsz; ++t) {
        const size_t bprev = base + (size_t)(t - 1) * U1;
        const size_t crow  = base + (size_t)t * U1;
        __builtin_prefetch(&blank[bprev + U1], 0, 1);   // global_prefetch_b8
        __builtin_prefetch(&cpre[crow + U1], 0, 1);

        float bp = blank[bprev + u];
        float cu = cpre[crow + u];
        float y  = (alpha + bp) - cu;

        sh[u] = y;
        __syncthreads();
        #pragma unroll
        for (int off = 1; off < 64; off <<= 1) {        // logaddexp scan
            float v = (u >= off) ? sh[u - off] : 0.0f;
            __syncthreads();
            if (u >= off) sh[u] = lae(sh[u], v);
            __syncthreads();
        }
        float L = sh[u];
        float anew = cu + L;

        if (u == 63) {                                  // u = 64 tail
            float bp64 = blank[bprev + 64];
            float c64  = cpre[crow + 64];
            float x64  = a64 + bp64;
            sh_a64 = c64 + lae(L, x64 - c64);
        }
        __syncthreads();
        a64 = sh_a64;

        if (t == t_idx) {
            if (tl < 64 && u == tl) ll[b] = anew + blank[crow + tl];
            if (tl == 64 && u == 63) ll[b] = a64 + blank[crow + 64];
        }
        alpha = anew;
    }
}

// ---------------------------------------------------------------------------
// Kernel 3: loss = -mean(ll). B == 32 == warpSize, one wave reduces it.
// ---------------------------------------------------------------------------
__global__ void rnnt_reduce(const float* __restrict__ ll, float* __restrict__ out) {
    float v = ll[threadIdx.x];
    #pragma unroll
    for (int off = 16; off > 0; off >>= 1) v += __shfl_down(v, off, 32);
    if (threadIdx.x == 0) out[0] = -v / (float)Bsz;
}

extern "C" void kernel_launch(void* const* d_in, const int* in_sizes, int n_in,
                              void* d_out, int out_size, void* d_ws, size_t ws_size,
                              hipStream_t stream) {
    (void)in_sizes; (void)n_in; (void)out_size; (void)ws_size;
    const float* lp      = (const float*)d_in[0];
    const int*   targets = (const int*)d_in[1];
    const int*   ilen    = (const int*)d_in[2];
    const int*   tlen    = (const int*)d_in[3];
    float* out = (float*)d_out;

    float* blank = (float*)d_ws;                        // B*T*65 floats
    float* cpre  = blank + (size_t)Bsz * Tsz * U1;      // B*T*65 floats
    float* ll    = cpre  + (size_t)Bsz * Tsz * U1;      // B floats

    rnnt_prep  <<<(Bsz * Tsz / 16) / 4, 128, 0, stream>>>(lp, targets, tlen, blank, cpre);
    rnnt_alpha <<<Bsz, 64, 0, stream>>>(blank, cpre, ilen, tlen, ll);
    rnnt_reduce<<<1, 32, 0, stream>>>(ll, out);
}